// ResponseCompute_38259568673285
// MI455X (gfx1250) — compile-verified
//
#include <hip/hip_runtime.h>
#include <hip/hip_bf16.h>

// ---------------------------------------------------------------------------
// ResponseCompute: depth-binned mean of bilinearly-upsampled feature maps,
// reformulated as   sums(64 x C) = Wmat(64 x P) * F(P x C)   and executed
// with V_WMMA_F32_16X16X4_F32 on gfx1250.
// ---------------------------------------------------------------------------

typedef __attribute__((ext_vector_type(2))) float v2f;
typedef __attribute__((ext_vector_type(8))) float v8f;

#define BATCH   4
#define DBINS   64
#define HOUT    256
#define WOUT    256
#define NPIX    (BATCH * HOUT * WOUT)      // 262144
#define C0      128
#define H0      64
#define W0DIM   64
#define P0      (BATCH * H0 * W0DIM)       // 16384
#define C1      256
#define H1      32
#define W1DIM   32
#define P1      (BATCH * H1 * W1DIM)       // 4096
#define KROWS   256
#define BIN_W   15.625f                    // 1000/64, exact in binary

// -------------------------- scatter: build Wmat ----------------------------

__device__ __forceinline__ void add_taps(float* __restrict__ Wm, int bin, int b,
                                         int y, int x, int hin, int win,
                                         float invscale) {
  float sy = (y + 0.5f) * invscale - 0.5f;
  float sx = (x + 0.5f) * invscale - 0.5f;
  int iy = (int)floorf(sy), ix = (int)floorf(sx);
  float fy = sy - (float)iy, fx = sx - (float)ix;
  int y0 = min(max(iy, 0), hin - 1);
  int y1 = min(max(iy + 1, 0), hin - 1);
  int x0 = min(max(ix, 0), win - 1);
  int x1 = min(max(ix + 1, 0), win - 1);
  int hw = hin * win;
  size_t base = (size_t)bin * (size_t)(BATCH * hw) + (size_t)b * hw;
  unsafeAtomicAdd(&Wm[base + y0 * win + x0], (1.f - fy) * (1.f - fx));
  unsafeAtomicAdd(&Wm[base + y0 * win + x1], (1.f - fy) * fx);
  unsafeAtomicAdd(&Wm[base + y1 * win + x0], fy * (1.f - fx));
  unsafeAtomicAdd(&Wm[base + y1 * win + x1], fy * fx);
}

__global__ void scatter_bins_kernel(const float* __restrict__ depths,
                                    float* __restrict__ Wm0,
                                    float* __restrict__ Wm1,
                                    float* __restrict__ counts) {
  __shared__ float lcnt[DBINS];
  int t = threadIdx.x;
  if (t < DBINS) lcnt[t] = 0.f;
  __syncthreads();

  int idx = blockIdx.x * blockDim.x + t;   // exact grid: NPIX threads
  float d = depths[idx];
  int bin = (int)floorf(d * (1.0f / BIN_W));
  // boundary fixup so result matches searchsorted(edges, d, 'right') - 1
  if (bin < DBINS - 1 && d >= (float)(bin + 1) * BIN_W) bin++;
  if (bin > 0 && d < (float)bin * BIN_W) bin--;
  bin = min(DBINS - 1, max(0, bin));

  atomicAdd(&lcnt[bin], 1.0f);

  int x = idx & (WOUT - 1);
  int y = (idx >> 8) & (HOUT - 1);
  int b = idx >> 16;

  add_taps(Wm0, bin, b, y, x, H0, W0DIM, 0.25f);    // 4x upsample
  add_taps(Wm1, bin, b, y, x, H1, W1DIM, 0.125f);   // 8x upsample

  __syncthreads();
  if (t < DBINS) unsafeAtomicAdd(&counts[t], lcnt[t]);
}

// ------------------- WMMA f32 GEMM: sums = Wmat * F ------------------------
// Wmat is (DBINS x P) row-major.  F(k, c) = fmap[b][c][y][x] with
// k = b*HW + y*w + x.  One wave per 16x16 (bin x channel) tile per K-split.

template <int C, int P, int HW, int LOG2HW, int KSPLIT>
__global__ void gemm_wmma_kernel(const float* __restrict__ Wmat,
                                 const float* __restrict__ fmap,
                                 float* __restrict__ sums) {
  constexpr int TILES_N = C / 16;
  constexpr int NASSIGN = 4 * TILES_N * KSPLIT;    // 4 M-tiles of bins
  constexpr int CHUNK = P / KSPLIT;                // multiple of 4

  int wave = (blockIdx.x * blockDim.x + threadIdx.x) >> 5;
  int lane = threadIdx.x & 31;
  if (wave >= NASSIGN) return;                     // wave-uniform exit

  int ks = wave % KSPLIT;
  int tile = wave / KSPLIT;
  int m0 = (tile & 3) * 16;                        // bin-tile
  int n0 = (tile >> 2) * 16;                       // channel-tile

  int row = lane & 15;                             // A: M row / B,C: N col
  int half = lane >> 4;                            // upper lanes: +2 in K (A/B)

  const float* arow = Wmat + (size_t)(m0 + row) * P + 2 * half;
  int col = n0 + row;                              // channel for this lane

  int k_begin = ks * CHUNK;
  int k_end = k_begin + CHUNK;

  v8f acc = {};
  for (int kk = k_begin; kk < k_end; kk += 4) {
    int k = kk + 2 * half;
    // A fragment: VGPR0=K(2*half), VGPR1=K(2*half+1), lanes 0-15 => M rows
    v2f a = *(const v2f*)(arow + kk);
    // B fragment: VGPR0=K(2*half), VGPR1=K(2*half+1), lanes 0-15 => N cols
    int bimg = k >> LOG2HW;
    int rem = k & (HW - 1);                        // k,k+1 stay in one image
    v2f b = *(const v2f*)(fmap + ((size_t)bimg * C + col) * HW + rem);
#if __has_builtin(__builtin_amdgcn_wmma_f32_16x16x4_f32)
    acc = __builtin_amdgcn_wmma_f32_16x16x4_f32(
        /*neg_a=*/false, a, /*neg_b=*/false, b,
        /*c_mod=*/(short)0, acc, /*reuse_a=*/false, /*reuse_b=*/false);
#else
    // fallback (wrong layout, compile-keeper only; histogram shows wmma=0)
    acc[0] += a.x * b.x + a.y * b.y;
#endif
  }

  // C/D layout: lane<16 -> M = v (+m0), lane>=16 -> M = v+8; N = lane&15
#pragma unroll
  for (int v = 0; v < 8; ++v) {
    int dbin = m0 + v + 8 * half;
    unsafeAtomicAdd(&sums[dbin * C + col], acc[v]);
  }
}

// ----------------------------- finalize ------------------------------------

__global__ void finalize_kernel(const float* __restrict__ sums0,
                                const float* __restrict__ sums1,
                                const float* __restrict__ counts,
                                float* __restrict__ out) {
  int i = blockIdx.x * blockDim.x + threadIdx.x;   // 0 .. 2*256*64-1
  int d = i & (DBINS - 1);
  int ch = (i >> 6) & (KROWS - 1);
  int layer = i >> 14;
  float cnt = counts[d];
  float val = 0.f;
  if (layer == 0) {
    if (ch < C0 && cnt > 0.f) val = sums0[d * C0 + ch] / cnt;
  } else {
    if (cnt > 0.f) val = sums1[d * C1 + ch] / cnt;
  }
  out[i] = val;
}

// ----------------------------- launcher ------------------------------------

extern "C" void kernel_launch(void* const* d_in, const int* in_sizes, int n_in,
                              void* d_out, int out_size, void* d_ws, size_t ws_size,
                              hipStream_t stream) {
  const float* fmap0 = (const float*)d_in[0];   // (4,128,64,64)
  const float* fmap1 = (const float*)d_in[1];   // (4,256,32,32)
  const float* depths = (const float*)d_in[2];  // (4,1,256,256)
  float* out = (float*)d_out;                   // (2,256,64)

  float* ws = (float*)d_ws;
  float* Wm0 = ws;                              // 64 * 16384
  float* Wm1 = Wm0 + (size_t)DBINS * P0;        // 64 * 4096
  float* counts = Wm1 + (size_t)DBINS * P1;     // 64
  float* sums0 = counts + DBINS;                // 64 * 128
  float* sums1 = sums0 + DBINS * C0;            // 64 * 256

  size_t zero_bytes =
      ((size_t)DBINS * P0 + (size_t)DBINS * P1 + DBINS +
       (size_t)DBINS * C0 + (size_t)DBINS * C1) * sizeof(float);
  hipMemsetAsync(d_ws, 0, zero_bytes, stream);

  scatter_bins_kernel<<<NPIX / 256, 256, 0, stream>>>(depths, Wm0, Wm1, counts);

  // layer0: 4 bin-tiles * 8 channel-tiles * 8 K-splits = 256 waves
  gemm_wmma_kernel<C0, P0, H0 * W0DIM, 12, 8>
      <<<(4 * (C0 / 16) * 8 * 32) / 256, 256, 0, stream>>>(Wm0, fmap0, sums0);
  // layer1: 4 * 16 * 8 = 512 waves
  gemm_wmma_kernel<C1, P1, H1 * W1DIM, 10, 8>
      <<<(4 * (C1 / 16) * 8 * 32) / 256, 256, 0, stream>>>(Wm1, fmap1, sums1);

  finalize_kernel<<<(2 * KROWS * DBINS) / 256, 256, 0, stream>>>(
      sums0, sums1, counts, out);
}